// ShiftConv1D_37271726195102
// MI455X (gfx1250) — compile-verified
//
#include <hip/hip_runtime.h>

// ShiftConv1D on MI455X. Output (32,16384,64) f32 = 128 MiB -> store-BW bound
// (~5.8us @ 23.3 TB/s). Width-3 conv as K=9 GEMM via V_WMMA_F32_16X16X4_F32
// (exact f32), K padded to 12 -> 3 chained WMMAs x 4 N-tiles = 12 WMMA/wave.
// Cross-conv (1x1, 3->3 + leaky 0.2) computed once per wave into a wave-private
// LDS slab; A fragments via ds_load at index 3m+k (zero slot for k>=9).
// Weights are zero-padded to K=12 in d_ws by a prep kernel so ALL B-fragment
// loads are unconditional (no divergent-exec load blocks in the WMMA tail).

#define SHIFT   10
#define ALPHA   0.2f
#define L_LEN   16384
#define BATCH   32
#define FOUT    64
#define KPAD    12
#define TILES_PER_BATCH (L_LEN / 16)   // 1024
#define WAVES_PER_BLOCK 8

typedef __attribute__((ext_vector_type(2))) float v2f;
typedef __attribute__((ext_vector_type(8))) float v8f;

__device__ __forceinline__ float leaky(float s) {
  // branch/select-free: max(s,0) + alpha*min(s,0)
  return fmaxf(s, 0.0f) + ALPHA * fminf(s, 0.0f);
}

// Zero-pad conv weights (3,3,64)=576 floats to (12,64)=768 floats in scratch.
__global__ __launch_bounds__(256)
void pad_weights(const float* __restrict__ wk, float* __restrict__ wpad) {
  const int i = blockIdx.x * blockDim.x + threadIdx.x;
  if (i < KPAD * FOUT) wpad[i] = (i < 9 * FOUT) ? wk[i] : 0.0f;
}

__global__ __launch_bounds__(256)
void shiftconv1d_wmma_f32(const float* __restrict__ x,      // (32,16384,1)
                          const float* __restrict__ ckg,    // (1,3,3)  [i][o]
                          const float* __restrict__ wpad,   // (12,64) zero-padded
                          float* __restrict__ out) {        // (32,16384,64)
  __shared__ float cross_lds[WAVES_PER_BLOCK][64];   // 2 KB/block, wave-private slabs

  const int lane = threadIdx.x & 31;
  const int widx = threadIdx.x >> 5;
  const int gw   = blockIdx.x * WAVES_PER_BLOCK + widx;
  const int b    = gw >> 10;                 // gw / TILES_PER_BATCH
  const int tile = gw & (TILES_PER_BATCH - 1);
  const int l0   = tile << 4;                // 16 output positions per wave

  const float* xb = x + (size_t)b * L_LEN;
  float* cl = cross_lds[widx];

  __builtin_prefetch(xb + l0, 0, 0);         // global_prefetch_b8 of the x window

  // Zero the K-padding slots (54..63) so A reads for k>=9 can hit slot 63.
  if (lane >= 22) cl[lane + 32] = 0.0f;

  // Lanes 0..17: cross at position l0-1+lane, ALL 3 channels (compile-time ck
  // indices -> scalar operands, no select trees). SAME padding: zero outside
  // [0,L); the +/-SHIFT taps are circular (jnp.roll).
  if (lane < 18) {
    const int pos = l0 - 1 + lane;
    float c0 = 0.0f, c1 = 0.0f, c2 = 0.0f;
    if (pos >= 0 && pos < L_LEN) {
      int pm = pos - SHIFT; if (pm < 0)      pm += L_LEN;
      int pp = pos + SHIFT; if (pp >= L_LEN) pp -= L_LEN;
      const float x0 = xb[pos], xm = xb[pm], xp = xb[pp];
      c0 = leaky(fmaf(xp, ckg[6], fmaf(xm, ckg[3], x0 * ckg[0])));
      c1 = leaky(fmaf(xp, ckg[7], fmaf(xm, ckg[4], x0 * ckg[1])));
      c2 = leaky(fmaf(xp, ckg[8], fmaf(xm, ckg[5], x0 * ckg[2])));
    }
    cl[3 * lane + 0] = c0;
    cl[3 * lane + 1] = c1;
    cl[3 * lane + 2] = c2;
  }
  // No barrier: slab is wave-private and LDS ops from one wave are in-order.

  const int m    = lane & 15;          // A row / B,C,D column
  const int half = (lane >> 4) & 1;

  const v8f zero8 = {0.f, 0.f, 0.f, 0.f, 0.f, 0.f, 0.f, 0.f};
  v8f acc[4];
#pragma unroll
  for (int t = 0; t < 4; ++t) acc[t] = zero8;

  // K-chain: k = w*3 + c in [0,9), padded to 12 -> 3 WMMA steps of K=4.
#pragma unroll
  for (int s = 0; s < 3; ++s) {
    // A fragment (16x4 f32): lane(half,m) element kk = A[m][2*half+kk],
    // which lives at LDS index (m+w)*3 + c == 3m + k.
    v2f a;
#pragma unroll
    for (int kk = 0; kk < 2; ++kk) {
      const int k   = 4 * s + 2 * half + kk;
      const int idx = (k <= 8) ? (3 * m + k) : 63;   // slot 63 holds 0.0f
      a[kk] = cl[idx];
    }
    // B fragments (4x16 f32): unconditional loads from the zero-padded weights.
#pragma unroll
    for (int t = 0; t < 4; ++t) {
      v2f bb;
#pragma unroll
      for (int kk = 0; kk < 2; ++kk) {
        const int k = 4 * s + 2 * half + kk;         // 0..11, always in-bounds
        bb[kk] = wpad[k * FOUT + t * 16 + m];
      }
      // 8 args: (neg_a, A, neg_b, B, c_mod, C, reuse_a, reuse_b)
      acc[t] = __builtin_amdgcn_wmma_f32_16x16x4_f32(
          false, a, false, bb, (short)0, acc[t], false, false);
    }
  }

  // D layout: VGPR r, lane(half,n): row = r + 8*half, col = n.
  // Lanes 0..15 store 64 consecutive channel floats per row -> coalesced.
  float* ob = out + ((size_t)b * L_LEN + l0) * FOUT;
#pragma unroll
  for (int t = 0; t < 4; ++t) {
#pragma unroll
    for (int r = 0; r < 8; ++r) {
      const int p = r + 8 * half;
      ob[(size_t)p * FOUT + t * 16 + m] = acc[t][r];
    }
  }
}

extern "C" void kernel_launch(void* const* d_in, const int* in_sizes, int n_in,
                              void* d_out, int out_size, void* d_ws, size_t ws_size,
                              hipStream_t stream) {
  (void)in_sizes; (void)n_in; (void)out_size; (void)ws_size;
  const float* x  = (const float*)d_in[0];   // 32*16384 floats
  const float* ck = (const float*)d_in[1];   // 9 floats
  const float* wk = (const float*)d_in[2];   // 576 floats (3,3,64)
  float* out  = (float*)d_out;               // 32*16384*64 floats
  float* wpad = (float*)d_ws;                // 768 floats (3 KB) of scratch

  // Pad weights (3,3,64) -> (12,64) with zero rows for k=9..11.
  pad_weights<<<3, 256, 0, stream>>>(wk, wpad);

  const int total_waves = BATCH * TILES_PER_BATCH;          // 32768
  const int blocks = total_waves / WAVES_PER_BLOCK;         // 4096
  shiftconv1d_wmma_f32<<<blocks, 256, 0, stream>>>(x, ck, wpad, out);
}